// GraphPoolingLayer_867583393845
// MI455X (gfx1250) — compile-verified
//
#include <hip/hip_runtime.h>
#include <hip/hip_bf16.h>

// ---------------------------------------------------------------------------
// TopKPooling (PyG-style) for MI455X / gfx1250, wave32.
//   B=16 graphs, N=8192 nodes/graph, D=256 features, K=4096, E=2,097,152.
// Pipeline:
//   1) score = tanh((x.w)/||w||)    -> WMMA F32 16x16x4 (native fp32 matvec)
//   2) per-graph top-K              -> 64KB-LDS bitonic sort, 1 WG per graph
//   3) x_out gather+scale, mapping scatter, edge filter -> bandwidth kernels
// Everything is HBM-bandwidth bound (~300 MB total traffic, ~13us @ 23.3TB/s).
// ---------------------------------------------------------------------------

typedef __attribute__((ext_vector_type(2))) float v2f;
typedef __attribute__((ext_vector_type(8))) float v8f;

#define GP_B 16
#define GP_N 8192
#define GP_D 256
#define GP_K 4096

// ---------------------------------------------------------------------------
// Kernel 1: scores via V_WMMA_F32_16X16X4_F32.
// One wave computes 16 rows. A-frag layout (ISA 7.12.2, 32-bit A 16x4):
//   lane L: row = L%16, holds K = {k0+2*(L/16), k0+1+2*(L/16)} as float2.
// B-frag (4x16) uses the same formula on w -> w replicated across all 16
// columns, so every result column equals the matvec. C layout: lane L,
// vgpr v -> row v + 8*(L/16); lanes 0 and 16 store the 16 scores.
// ---------------------------------------------------------------------------
__global__ __launch_bounds__(256) void gp_score_wmma_kernel(
    const float* __restrict__ x, const float* __restrict__ w,
    float* __restrict__ scores) {
  const int lane = threadIdx.x & 31;
  const int wave = (blockIdx.x * blockDim.x + threadIdx.x) >> 5;
  const int r0 = wave * 16;

  // ||w||: D=256 -> each lane squares 8 elements, wave32 xor-reduction.
  float s = 0.f;
#pragma unroll
  for (int t = 0; t < 8; ++t) {
    float v = w[lane * 8 + t];
    s += v * v;
  }
#pragma unroll
  for (int off = 16; off > 0; off >>= 1) s += __shfl_xor(s, off, 32);
  const float invnorm = rsqrtf(s);

  const int half = lane >> 4;       // 0: K pair {0,1}, 1: K pair {2,3}
  const int rowInTile = lane & 15;
  const long rowBase = (long)(r0 + rowInTile) * GP_D;

  v8f c = {0.f, 0.f, 0.f, 0.f, 0.f, 0.f, 0.f, 0.f};
#pragma unroll 8
  for (int k0 = 0; k0 < GP_D; k0 += 4) {
    v2f a = *(const v2f*)(x + rowBase + k0 + 2 * half);
    v2f b = *(const v2f*)(w + k0 + 2 * half);
    // 8 args: (neg_a, A, neg_b, B, c_mod, C, reuse_a, reuse_b)
    c = __builtin_amdgcn_wmma_f32_16x16x4_f32(false, a, false, b, (short)0, c,
                                              false, false);
  }

  if (rowInTile == 0) {  // lanes 0 and 16 hold rows r0..r0+7 / r0+8..r0+15
#pragma unroll
    for (int j = 0; j < 8; ++j)
      scores[r0 + half * 8 + j] = tanhf(c[j] * invnorm);
  }
}

// ---------------------------------------------------------------------------
// Kernel 2: per-graph top-K via full bitonic sort of 8192 u64 keys in LDS.
// key = sortable(scoreBits) << 32 | (N-1-idx): descending score, ties by
// ascending original index (matches lax.top_k ordering).
// ---------------------------------------------------------------------------
__global__ __launch_bounds__(1024) void gp_topk_sort_kernel(
    const float* __restrict__ scores, int* __restrict__ perm,
    float* __restrict__ vals) {
  __shared__ unsigned long long keys[GP_N];  // 64 KB LDS
  const int g = blockIdx.x;
  const int tid = threadIdx.x;
  const float* s = scores + g * GP_N;

  for (int i = tid; i < GP_N; i += 1024) {
    unsigned int bits = __float_as_uint(s[i]);
    bits = (bits & 0x80000000u) ? ~bits : (bits | 0x80000000u);
    keys[i] = ((unsigned long long)bits << 32) |
              (unsigned int)(GP_N - 1 - i);
  }
  __syncthreads();

  for (int k = 2; k <= GP_N; k <<= 1) {
    for (int j = k >> 1; j > 0; j >>= 1) {
      for (int i = tid; i < GP_N; i += 1024) {
        int ixj = i ^ j;
        if (ixj > i) {
          bool desc = ((i & k) == 0);
          unsigned long long a = keys[i];
          unsigned long long b = keys[ixj];
          bool doswap = desc ? (a < b) : (a > b);
          if (doswap) {
            keys[i] = b;
            keys[ixj] = a;
          }
        }
      }
      __syncthreads();
    }
  }

  for (int i = tid; i < GP_K; i += 1024) {
    int local = GP_N - 1 - (int)(unsigned int)(keys[i] & 0xFFFFFFFFu);
    perm[g * GP_K + i] = g * GP_N + local;
    vals[g * GP_K + i] = s[local];
  }
}

// ---------------------------------------------------------------------------
// Kernel 3: mapping init (-1 everywhere).
// ---------------------------------------------------------------------------
__global__ void gp_init_mapping_kernel(int* __restrict__ mapping, int n) {
  int i = blockIdx.x * blockDim.x + threadIdx.x;
  if (i < n) mapping[i] = -1;
}

// ---------------------------------------------------------------------------
// Kernel 4: scatter new ids + batch_out (batch[global] == global / N).
// ---------------------------------------------------------------------------
__global__ void gp_scatter_kernel(const int* __restrict__ perm,
                                  int* __restrict__ mapping,
                                  float* __restrict__ batch_out, int nKept) {
  int i = blockIdx.x * blockDim.x + threadIdx.x;
  if (i < nKept) {
    int p = perm[i];
    mapping[p] = i;
    batch_out[i] = (float)(p / GP_N);
  }
}

// ---------------------------------------------------------------------------
// Kernel 5: x_out[i] = x[perm[i]] * vals[i], float4 per thread (128-bit).
// ---------------------------------------------------------------------------
__global__ __launch_bounds__(256) void gp_gather_scale_kernel(
    const float* __restrict__ x, const int* __restrict__ perm,
    const float* __restrict__ vals, float* __restrict__ xout, int nKept) {
  int t = blockIdx.x * blockDim.x + threadIdx.x;
  int row = t >> 6;  // D/4 = 64 quads per row
  int q = t & 63;
  if (row < nKept) {
    const float4* src = (const float4*)(x + (long)perm[row] * GP_D);
    float4 v = src[q];
    float sc = vals[row];
    float4 o = make_float4(v.x * sc, v.y * sc, v.z * sc, v.w * sc);
    ((float4*)(xout + (long)row * GP_D))[q] = o;
  }
}

// ---------------------------------------------------------------------------
// Kernel 6: edge filter + reindex. Dropped edges -> -1 in both rows.
// ---------------------------------------------------------------------------
__global__ void gp_edge_kernel(const int* __restrict__ ei,
                               const int* __restrict__ mapping,
                               float* __restrict__ eout, int E) {
  int e = blockIdx.x * blockDim.x + threadIdx.x;
  if (e < E) {
    int ns = mapping[ei[e]];
    int nd = mapping[ei[E + e]];
    bool kept = (ns >= 0) && (nd >= 0);
    eout[e] = kept ? (float)ns : -1.0f;
    eout[E + e] = kept ? (float)nd : -1.0f;
  }
}

// ---------------------------------------------------------------------------
extern "C" void kernel_launch(void* const* d_in, const int* in_sizes, int n_in,
                              void* d_out, int out_size, void* d_ws,
                              size_t ws_size, hipStream_t stream) {
  const float* x = (const float*)d_in[0];       // [B*N, D]
  const int* edge_index = (const int*)d_in[1];  // [2, E]
  // d_in[2] = batch (unused; derived as global/N)
  const float* weight = (const float*)d_in[3];  // [D]

  const int totalNodes = GP_B * GP_N;  // 131072
  const int nKept = GP_B * GP_K;       // 65536
  const int E = in_sizes[1] / 2;       // 2097152

  // Workspace layout (bytes): scores | perm | vals | mapping
  char* ws = (char*)d_ws;
  float* scores = (float*)ws;                          // 512 KB
  int* perm = (int*)(ws + (size_t)totalNodes * 4);     // 256 KB
  float* vals = (float*)(ws + (size_t)totalNodes * 4 + (size_t)nKept * 4);
  int* mapping = (int*)(ws + (size_t)totalNodes * 4 + (size_t)nKept * 8);

  // Output layout (float32): x_out | new_edge_index | batch_out
  float* out_x = (float*)d_out;
  float* out_edges = out_x + (size_t)nKept * GP_D;
  float* out_batch = out_edges + (size_t)2 * E;

  // 1) scores: 8192 waves of 16 rows; 8 waves / 256-thread block.
  {
    int waves = totalNodes / 16;
    int blocks = waves / 8;
    gp_score_wmma_kernel<<<blocks, 256, 0, stream>>>(x, weight, scores);
  }
  // 2) top-K per graph.
  gp_topk_sort_kernel<<<GP_B, 1024, 0, stream>>>(scores, perm, vals);
  // 3) mapping = -1.
  gp_init_mapping_kernel<<<(totalNodes + 255) / 256, 256, 0, stream>>>(
      mapping, totalNodes);
  // 4) scatter new ids + batch_out.
  gp_scatter_kernel<<<(nKept + 255) / 256, 256, 0, stream>>>(perm, mapping,
                                                             out_batch, nKept);
  // 5) x_out gather+scale (float4 per thread).
  {
    long threads = (long)nKept * (GP_D / 4);
    int blocks = (int)((threads + 255) / 256);
    gp_gather_scale_kernel<<<blocks, 256, 0, stream>>>(x, perm, vals, out_x,
                                                       nKept);
  }
  // 6) edge filter.
  gp_edge_kernel<<<(E + 255) / 256, 256, 0, stream>>>(edge_index, mapping,
                                                      out_edges, E);
}